// GHM_Loss_70970039599231
// MI455X (gfx1250) — compile-verified
//
#include <hip/hip_runtime.h>

// ---------------------------------------------------------------------------
// GHM loss, gfx1250 (MI455X).  Bandwidth-bound: 2 streaming passes over
// x/target (~536 MB => ~23 us @ 23.3 TB/s).  Histogram (13.4 MB) is
// L2-resident (192 MB L2).  Final scalar reduction runs through
// V_WMMA_F32_16X16X4_F32 (B = ones => D[m,n] = sum_k A[m,k] + C[m,n]).
// Round 2: fast-path transcendentals (v_log_f32 / v_rcp_f32) + float4 loss
// pass so VALU time drops well below the HBM streaming time.
// ---------------------------------------------------------------------------

typedef __attribute__((ext_vector_type(2))) float v2f;
typedef __attribute__((ext_vector_type(8))) float v8f;

#define GHM_N     33554432     // 4096 * 8192
#define GHM_NBIN  3355443      // N / BINS, BINS = 10
// ws layout
#define WS_NZ_OFF   ((size_t)GHM_NBIN * 4)        // 13,421,772
#define WS_ACC_OFF  (WS_NZ_OFF + 4)               // 13,421,776 (8-aligned)

__device__ __forceinline__ int ghm_bin(float g) {
    const float SCALE = (float)((double)GHM_NBIN - 1e-4);  // matches jnp f32 math
    int b = (int)floorf(g * SCALE);
    return (b > GHM_NBIN - 1) ? (GHM_NBIN - 1) : b;
}

// fast natural log: v_log_f32 (log2) * ln2  -- ~2 VALU ops
__device__ __forceinline__ float fast_ln(float v) {
    return __builtin_amdgcn_logf(v) * 0.69314718055994530942f;
}

// weighted BCE term for one element
__device__ __forceinline__ float ghm_term(float xv, float tv,
                                          const unsigned int* __restrict__ hist,
                                          float nzf, float Nf) {
    int   b = ghm_bin(fabsf(xv - tv));
    // w = N / max(count*nz, 1)  via single v_rcp_f32
    float w = Nf * __builtin_amdgcn_rcpf(fmaxf((float)hist[b] * nzf, 1.0f));
    float l = -(tv * fast_ln(xv) + (1.0f - tv) * fast_ln(1.0f - xv));
    return l * w;
}

// ---- pass 0: zero the workspace -------------------------------------------
__global__ void ghm_zero(unsigned int* __restrict__ hist,
                         unsigned int* __restrict__ nz,
                         double*       __restrict__ acc) {
    unsigned i = blockIdx.x * blockDim.x + threadIdx.x;
    unsigned stride = gridDim.x * blockDim.x;
    for (unsigned j = i; j < GHM_NBIN; j += stride) hist[j] = 0u;
    if (i == 0) { *nz = 0u; *acc = 0.0; }
}

// ---- pass 1: histogram of gradient-norm bins (float4 streaming) -----------
__global__ void ghm_hist(const float4* __restrict__ x,
                         const float4* __restrict__ t,
                         unsigned int* __restrict__ hist) {
    unsigned i = blockIdx.x * blockDim.x + threadIdx.x;
    unsigned stride = gridDim.x * blockDim.x;
    const unsigned NQ = GHM_N / 4;
    for (unsigned q = i; q < NQ; q += stride) {
        // stream-ahead prefetch (global_prefetch_b8 on gfx1250)
        if (q + stride < NQ) {
            __builtin_prefetch(&x[q + stride], 0, 0);
            __builtin_prefetch(&t[q + stride], 0, 0);
        }
        float4 xv = x[q];
        float4 tv = t[q];
        atomicAdd(&hist[ghm_bin(fabsf(xv.x - tv.x))], 1u);
        atomicAdd(&hist[ghm_bin(fabsf(xv.y - tv.y))], 1u);
        atomicAdd(&hist[ghm_bin(fabsf(xv.z - tv.z))], 1u);
        atomicAdd(&hist[ghm_bin(fabsf(xv.w - tv.w))], 1u);
    }
}

// ---- pass 2: count non-empty bins -----------------------------------------
__global__ void ghm_nz(const unsigned int* __restrict__ hist,
                       unsigned int* __restrict__ nz) {
    unsigned i = blockIdx.x * blockDim.x + threadIdx.x;
    unsigned stride = gridDim.x * blockDim.x;
    unsigned local = 0u;
    for (unsigned j = i; j < GHM_NBIN; j += stride)
        local += (hist[j] > 0u) ? 1u : 0u;
    // wave32 tree reduction
    for (int off = 16; off > 0; off >>= 1)
        local += __shfl_down(local, off, 32);
    if ((threadIdx.x & 31u) == 0u && local != 0u)
        atomicAdd(nz, local);
}

// ---- pass 3: fused weight-gather + weighted BCE + WMMA reduction ----------
// Each lane folds 4 fp32 loss terms per iteration into the wave's 16x4 f32
// A-tile (pair-summed into 2 VGPRs); B = ones so C accumulates every term
// (x16 replication across columns, divided out in the finalize step).
// 2048*256 threads over N/4 quads => exactly 16 uniform iterations, EXEC
// all-ones at every WMMA as the ISA requires.
__global__ void ghm_loss(const float4* __restrict__ x,
                         const float4* __restrict__ t,
                         const unsigned int* __restrict__ hist,
                         const unsigned int* __restrict__ nzp,
                         double* __restrict__ acc) {
    const float nzf = (float)(*nzp);
    const float Nf  = (float)GHM_N;

    unsigned gtid   = blockIdx.x * blockDim.x + threadIdx.x;
    unsigned stride = gridDim.x * blockDim.x;
    const unsigned NQ = GHM_N / 4;   // 8,388,608 ; 16 * (2048*256)

    v8f c = {};                       // f32 C accumulator (8 VGPRs)
    v2f ones; ones[0] = 1.0f; ones[1] = 1.0f;

    for (unsigned q = gtid; q < NQ; q += stride) {
        if (q + stride < NQ) {
            __builtin_prefetch(&x[q + stride], 0, 0);
            __builtin_prefetch(&t[q + stride], 0, 0);
        }
        float4 xv = x[q];
        float4 tv = t[q];

        float l0 = ghm_term(xv.x, tv.x, hist, nzf, Nf);
        float l1 = ghm_term(xv.y, tv.y, hist, nzf, Nf);
        float l2 = ghm_term(xv.z, tv.z, hist, nzf, Nf);
        float l3 = ghm_term(xv.w, tv.w, hist, nzf, Nf);

        v2f a; a[0] = l0 + l1; a[1] = l2 + l3;
        // D = A(16x4 f32) * ones(4x16) + C  => rowsum accumulate
        c = __builtin_amdgcn_wmma_f32_16x16x4_f32(
                /*neg_a=*/false, a, /*neg_b=*/false, ones,
                /*c_mod=*/(short)0, c, /*reuse_a=*/false, /*reuse_b=*/false);
    }

    float lane = c[0] + c[1] + c[2] + c[3] + c[4] + c[5] + c[6] + c[7];

    __shared__ float red[256];
    red[threadIdx.x] = lane;
    __syncthreads();
    for (int s = 128; s > 0; s >>= 1) {
        if ((int)threadIdx.x < s) red[threadIdx.x] += red[threadIdx.x + s];
        __syncthreads();
    }
    if (threadIdx.x == 0)
        atomicAdd(acc, (double)red[0]);
}

// ---- pass 4: finalize mean ------------------------------------------------
__global__ void ghm_final(const double* __restrict__ acc,
                          float* __restrict__ out) {
    if (blockIdx.x == 0 && threadIdx.x == 0)
        out[0] = (float)(*acc / (16.0 * (double)GHM_N));  // /16: WMMA column replication
}

// ---------------------------------------------------------------------------
extern "C" void kernel_launch(void* const* d_in, const int* in_sizes, int n_in,
                              void* d_out, int out_size, void* d_ws, size_t ws_size,
                              hipStream_t stream) {
    (void)in_sizes; (void)n_in; (void)out_size; (void)ws_size;

    const float* x = (const float*)d_in[0];
    const float* t = (const float*)d_in[1];
    float* out = (float*)d_out;

    unsigned int* hist = (unsigned int*)d_ws;
    unsigned int* nz   = (unsigned int*)((char*)d_ws + WS_NZ_OFF);
    double*       acc  = (double*)((char*)d_ws + WS_ACC_OFF);

    ghm_zero <<<1024, 256, 0, stream>>>(hist, nz, acc);
    ghm_hist <<<2048, 256, 0, stream>>>((const float4*)x, (const float4*)t, hist);
    ghm_nz   <<<1024, 256, 0, stream>>>(hist, nz);
    ghm_loss <<<2048, 256, 0, stream>>>((const float4*)x, (const float4*)t,
                                        hist, nz, acc);
    ghm_final<<<1, 1, 0, stream>>>(acc, out);
}